// DeepFeatureLoss_12850542149713
// MI455X (gfx1250) — compile-verified
//
#include <hip/hip_runtime.h>
#include <hip/hip_bf16.h>

typedef float v2f __attribute__((ext_vector_type(2)));
typedef float v8f __attribute__((ext_vector_type(8)));

#define SIGMA_INV 200.0f   // 1 / 0.005

// ---------------- kernel 0: zero the outputs ----------------
__global__ void dfl_zero(float* out, int n) {
    int i = blockIdx.x * blockDim.x + threadIdx.x;
    if (i < n) out[i] = 0.0f;
}

// ---------------- kernel 1: row norms + reg loss ----------------
// f1n[i] = ||fea1_i||^2, f2n[i] = ||fea2_i||^2, pn[i] = ||p_i/sigma||^2
// reg[b] += (f1n + f2n) / (N*D)
__global__ void dfl_prep(const float* __restrict__ points,
                         const float* __restrict__ f1,
                         const float* __restrict__ f2,
                         float* __restrict__ f1n, float* __restrict__ f2n,
                         float* __restrict__ pn,
                         float* __restrict__ out,
                         int B, int N, int D) {
    int i = blockIdx.x * blockDim.x + threadIdx.x;
    if (i >= B * N) return;
    int b = i / N;
    const float* a = f1 + (size_t)i * D;
    const float* c = f2 + (size_t)i * D;
    float s1 = 0.f, s2 = 0.f;
    for (int d = 0; d < D; ++d) { s1 += a[d] * a[d]; s2 += c[d] * c[d]; }
    f1n[i] = s1; f2n[i] = s2;
    const float* p = points + (size_t)i * 3;
    float x = p[0] * SIGMA_INV, y = p[1] * SIGMA_INV, z = p[2] * SIGMA_INV;
    pn[i] = x * x + y * y + z * z;
    atomicAdd(&out[B + b], (s1 + s2) / (float)(N * D));
}

// ---------------- kernel 2: fused flash-style CE loss ----------------
// One wave32 per (batch, 16-row block). Streams over all 256 column tiles,
// computing both gram tiles with V_WMMA_F32_16X16X4_F32 and keeping online
// softmax statistics in registers. Never materializes anything N x N.
__global__ void __launch_bounds__(32)
dfl_ce(const float* __restrict__ points,
       const float* __restrict__ f1,
       const float* __restrict__ f2,
       const float* __restrict__ w,
       const float* __restrict__ f1n,
       const float* __restrict__ f2n,
       const float* __restrict__ pn,
       float* __restrict__ out,
       int B, int N, int D) {
    const int lane   = threadIdx.x;        // 0..31
    const int hi     = lane >> 4;          // 0: K=0,1 half | 1: K=2,3 half
    const int lanelo = lane & 15;
    const int nblk   = N >> 4;
    const int b      = blockIdx.x / nblk;
    const int row0   = (blockIdx.x % nblk) << 4;

    // ---- Preload A tile (fea1 rows row0..row0+15), invariant over column sweep
    const int    arow   = row0 + lanelo;
    const size_t f1base = ((size_t)b * N + arow) * D;
    v2f a_reg[16];
#pragma unroll
    for (int kk = 0; kk < 16; ++kk) {
        const int k = kk * 4 + 2 * hi;                 // 8-byte aligned
        a_reg[kk] = *(const v2f*)(f1 + f1base + k);
    }
    // point A tile (K padded 3 -> 4 with zero), scaled by 1/sigma
    const size_t pbase = ((size_t)b * N + arow) * 3;
    v2f pa;
    pa[0] = points[pbase + 2 * hi] * SIGMA_INV;
    pa[1] = hi ? 0.0f : points[pbase + 1] * SIGMA_INV;

    // per-m constants: this lane's 8 accumulator rows are m = v + 8*hi
    float f1n_m[8], pn_m[8];
#pragma unroll
    for (int v = 0; v < 8; ++v) {
        const int m = v + 8 * hi;
        f1n_m[v] = f1n[(size_t)b * N + row0 + m];
        pn_m[v]  = pn [(size_t)b * N + row0 + m];
    }

    // online stats: s_p = sum exp(pd) (max fixed at 0 since pd<=0, pd_ii=0)
    //               t   = sum exp(pd)*fd
    //               (m_f, s_f) online max/sum for the feature log-softmax
    float s_p[8], t[8], m_f[8], s_f[8];
#pragma unroll
    for (int v = 0; v < 8; ++v) { s_p[v] = 0.f; t[v] = 0.f; m_f[v] = -3.0e38f; s_f[v] = 0.f; }

    for (int col0 = 0; col0 < N; col0 += 16) {
        const int    brow   = col0 + lanelo;
        const size_t f2base = ((size_t)b * N + brow) * D;

        // ---- feature gram tile: 16 chained fp32 WMMAs (K = 64)
        v8f acc = {};
#pragma unroll
        for (int kk = 0; kk < 16; ++kk) {
            const int k = kk * 4 + 2 * hi;
            v2f breg = *(const v2f*)(f2 + f2base + k);
            acc = __builtin_amdgcn_wmma_f32_16x16x4_f32(
                false, a_reg[kk], false, breg, (short)0, acc, false, false);
        }

        // ---- point gram tile: single fp32 WMMA (K = 4, padded)
        const size_t pb2 = ((size_t)b * N + brow) * 3;
        v2f pb;
        pb[0] = points[pb2 + 2 * hi] * SIGMA_INV;
        pb[1] = hi ? 0.0f : points[pb2 + 1] * SIGMA_INV;
        v8f pacc = {};
        pacc = __builtin_amdgcn_wmma_f32_16x16x4_f32(
            false, pa, false, pb, (short)0, pacc, false, false);

        const float f2n_j = f2n[(size_t)b * N + brow];
        const float pn_j  = pn [(size_t)b * N + brow];

        // ---- online softmax update for the 8 (m, n=col0+lanelo) pairs
#pragma unroll
        for (int v = 0; v < 8; ++v) {
            const float fd = 2.0f * acc[v]  - f1n_m[v] - f2n_j;  // -||f1_i - f2_j||^2
            const float pd = 2.0f * pacc[v] - pn_m[v]  - pn_j;   // -||pi - pj||^2/s^2
            const float ep = __expf(fminf(pd, 0.0f));
            s_p[v] += ep;
            t[v]   += ep * fd;
            const float nm = fmaxf(m_f[v], fd);
            s_f[v] = s_f[v] * __expf(m_f[v] - nm) + __expf(fd - nm);
            m_f[v] = nm;
        }
    }

    // ---- merge stats across the 16 lanes of each half-wave (butterfly)
#pragma unroll
    for (int v = 0; v < 8; ++v) {
#pragma unroll
        for (int off = 8; off > 0; off >>= 1) {
            const float o_m  = __shfl_xor(m_f[v], off, 16);
            const float o_s  = __shfl_xor(s_f[v], off, 16);
            const float o_sp = __shfl_xor(s_p[v], off, 16);
            const float o_t  = __shfl_xor(t[v],   off, 16);
            const float nm   = fmaxf(m_f[v], o_m);
            s_f[v] = s_f[v] * __expf(m_f[v] - nm) + o_s * __expf(o_m - nm);
            m_f[v] = nm;
            s_p[v] += o_sp;
            t[v]   += o_t;
        }
    }

    // ce_i = LSE_f(i) - (sum_j g_ij * fd_ij);  weighted sum over this block's rows
    if (lanelo == 0) {
        float sum = 0.f;
#pragma unroll
        for (int v = 0; v < 8; ++v) {
            const int m  = v + 8 * hi;
            const float ce = (m_f[v] + __logf(s_f[v])) - t[v] / s_p[v];
            sum += ce * w[(size_t)b * N + row0 + m];
        }
        atomicAdd(&out[b], sum);
    }
}

// ---------------- launcher ----------------
extern "C" void kernel_launch(void* const* d_in, const int* in_sizes, int n_in,
                              void* d_out, int out_size, void* d_ws, size_t ws_size,
                              hipStream_t stream) {
    const float* points = (const float*)d_in[0];
    const float* fea1   = (const float*)d_in[1];
    const float* fea2   = (const float*)d_in[2];
    const float* wts    = (const float*)d_in[3];
    float* out = (float*)d_out;

    const int B  = out_size / 2;          // outputs: ce_loss[B] ++ reg[B]
    const int BN = in_sizes[3];           // weights: B*N
    const int N  = BN / B;
    const int D  = in_sizes[1] / BN;

    float* ws  = (float*)d_ws;
    float* f1n = ws;
    float* f2n = ws + BN;
    float* pn  = ws + 2 * (size_t)BN;

    dfl_zero<<<1, 64, 0, stream>>>(out, out_size);
    dfl_prep<<<(BN + 255) / 256, 256, 0, stream>>>(points, fea1, fea2,
                                                   f1n, f2n, pn, out, B, N, D);
    dfl_ce<<<B * (N / 16), 32, 0, stream>>>(points, fea1, fea2, wts,
                                            f1n, f2n, pn, out, B, N, D);
}